// ModelNew_25056839205107
// MI455X (gfx1250) — compile-verified
//
#include <hip/hip_runtime.h>

// Exclusive row-wise scan, B=2048 rows x N=32768 fp32.
// Memory-bound: 512 MB total traffic -> ~22 us floor @ 23.3 TB/s.
// Strategy: one workgroup per row, stage the whole 128 KB row in LDS via
// gfx1250 async global->LDS copies (non-temporal hints: streaming data,
// never re-read, don't pollute L2), scan hierarchically (wave32 shuffles),
// stream back with async LDS->global stores. Exactly 1 read + 1 write of HBM.

#define ROW_N      32768
#define THREADS    1024
#define PER_THREAD 32            // floats per thread chunk (8 x float4)
#define CHUNK_PAD  36            // padded chunk stride in floats (144 B, 16B aligned)
#define LDS_BYTES  (THREADS * CHUNK_PAD * 4)   // 147456 B < 320 KB WGP LDS

__device__ __forceinline__ unsigned lds_byte_addr(const void* p) {
    // Generic (flat) LDS pointers carry the LDS byte offset in addr[31:0]
    // (CDNA5 flat->LDS aperture mapping). That is exactly what the async
    // instructions' LDS-address VGPR expects.
    return (unsigned)(size_t)p;
}

__device__ __forceinline__ void async_load_b128(unsigned lds_off, const void* gaddr) {
    asm volatile("global_load_async_to_lds_b128 %0, %1, off th:TH_LOAD_NT"
                 :: "v"(lds_off), "v"(gaddr) : "memory");
}

__device__ __forceinline__ void async_store_b128(void* gaddr, unsigned lds_off) {
    asm volatile("global_store_async_from_lds_b128 %0, %1, off th:TH_STORE_NT"
                 :: "v"(gaddr), "v"(lds_off) : "memory");
}

__device__ __forceinline__ void wait_async0() {
    asm volatile("s_wait_asynccnt 0" ::: "memory");
}

extern "C" __global__ void __launch_bounds__(THREADS, 1)
row_exclusive_scan(const float* __restrict__ x, float* __restrict__ y) {
    extern __shared__ float lds[];           // THREADS * CHUNK_PAD floats
    __shared__ float wave_base[32];

    const int tid  = threadIdx.x;
    const int lane = tid & 31;
    const int wid  = tid >> 5;

    const float* xr = x + (size_t)blockIdx.x * ROW_N;
    float*       yr = y + (size_t)blockIdx.x * ROW_N;

    const unsigned lds_base = lds_byte_addr(lds);

    // ---- Phase A: async copy row -> LDS -------------------------------
    // Global side coalesced: instr j, thread t moves float4 at element
    // ge = j*4096 + 4t (each wave = 512 contiguous bytes). LDS side lands
    // it in the padded per-thread chunk layout.
    #pragma unroll
    for (int j = 0; j < 8; ++j) {
        const int ge = j * 4096 + tid * 4;               // float index in row
        const unsigned c = (unsigned)ge >> 5;            // owning chunk
        const unsigned o = (unsigned)ge & 31u;           // offset in chunk
        const unsigned laddr = lds_base + c * (CHUNK_PAD * 4) + o * 4;
        async_load_b128(laddr, xr + ge);
    }
    wait_async0();
    __syncthreads();

    // ---- Phase B: per-thread chunk sum --------------------------------
    const float4* chunk4 = (const float4*)(lds + tid * CHUNK_PAD);
    float tsum = 0.0f;
    #pragma unroll
    for (int k = 0; k < 8; ++k) {
        const float4 v = chunk4[k];
        tsum += v.x; tsum += v.y; tsum += v.z; tsum += v.w;
    }

    // ---- Phase C: block-wide exclusive scan of the 1024 thread sums ---
    // Wave-level inclusive scan (wave32, 5 shuffle steps).
    float inc = tsum;
    #pragma unroll
    for (int d = 1; d < 32; d <<= 1) {
        const float n = __shfl_up(inc, (unsigned)d, 32);
        if (lane >= d) inc += n;
    }
    float lane_excl = __shfl_up(inc, 1u, 32);
    if (lane == 0) lane_excl = 0.0f;
    if (lane == 31) wave_base[wid] = inc;                // wave total
    __syncthreads();

    if (wid == 0) {                                      // scan the 32 wave totals
        const float w = wave_base[lane];
        float s = w;
        #pragma unroll
        for (int d = 1; d < 32; d <<= 1) {
            const float n = __shfl_up(s, (unsigned)d, 32);
            if (lane >= d) s += n;
        }
        float e = __shfl_up(s, 1u, 32);
        if (lane == 0) e = 0.0f;
        wave_base[lane] = e;                             // exclusive wave base
    }
    __syncthreads();

    float run = wave_base[wid] + lane_excl;              // thread's exclusive base

    // ---- Phase D: exclusive scan within the chunk, in place in LDS ----
    float4* chunkw = (float4*)(lds + tid * CHUNK_PAD);
    #pragma unroll
    for (int k = 0; k < 8; ++k) {
        const float4 in = chunkw[k];
        float4 out;
        out.x = run; run += in.x;
        out.y = run; run += in.y;
        out.z = run; run += in.z;
        out.w = run; run += in.w;
        chunkw[k] = out;
    }
    __syncthreads();

    // ---- Phase E: async store LDS -> global (coalesced, non-temporal) -
    #pragma unroll
    for (int j = 0; j < 8; ++j) {
        const int ge = j * 4096 + tid * 4;
        const unsigned c = (unsigned)ge >> 5;
        const unsigned o = (unsigned)ge & 31u;
        const unsigned laddr = lds_base + c * (CHUNK_PAD * 4) + o * 4;
        async_store_b128(yr + ge, laddr);
    }
    wait_async0();   // (S_ENDPGM also implies wait-idle; explicit for clarity)
}

extern "C" void kernel_launch(void* const* d_in, const int* in_sizes, int n_in,
                              void* d_out, int out_size, void* d_ws, size_t ws_size,
                              hipStream_t stream) {
    (void)n_in; (void)d_ws; (void)ws_size; (void)out_size;
    const float* x = (const float*)d_in[0];
    float*       y = (float*)d_out;
    const int rows = in_sizes[0] / ROW_N;   // 2048
    dim3 grid((unsigned)rows), block(THREADS);
    hipLaunchKernelGGL(row_exclusive_scan, grid, block, LDS_BYTES, stream, x, y);
}